// NewLawinHead_54683523613017
// MI455X (gfx1250) — compile-verified
//
#include <hip/hip_runtime.h>

typedef float v2f __attribute__((ext_vector_type(2)));
typedef float v8f __attribute__((ext_vector_type(8)));

#define BN_INV 0.9999950000374997f   /* 1/sqrt(1+1e-5) */
#define EPS_LN 1e-5f

// ---------------------------------------------------------------------------
// Batched GEMM via V_WMMA_F32_16X16X4_F32.
// Y[b][o][p] = act( (sum_c W[o][c] * X[b][c][p]) * (g?g[o]*gmul:1) + (bv?bv[o]:0) )
// One wave (block=32) computes a 16(M) x 64(N) strip: 4 accumulators share the
// A fragment each K-step. Requirements: P % 64 == 0 (true for all uses here);
// O may have a tail (clamped A-row loads; un-stored rows absorb garbage).
// Inner loop is branch-free: EXEC stays full for WMMA, pure pointer increments.
// ---------------------------------------------------------------------------
__global__ void gemm_wmma_f32(const float* __restrict__ W,
                              const float* __restrict__ X,
                              float* __restrict__ Y,
                              const float* __restrict__ g,
                              const float* __restrict__ bv,
                              float gmul, int relu,
                              int O, int C, int P,
                              long long xStride, long long yStride)
{
    const int lane = threadIdx.x & 31;
    const int nb0  = blockIdx.x * 64;              // N strip base
    const int mb   = blockIdx.y * 16;              // M tile base

    const float* Xb = X + (long long)blockIdx.z * xStride;
    float*       Yb = Y + (long long)blockIdx.z * yStride;

    const int l16 = lane & 15;
    const int hi  = lane >> 4;                     // lane half selects K pair
    int arow = mb + l16;
    if (arow > O - 1) arow = O - 1;                // clamp: rows >= O never stored

    // A: lane half reads (K+2*hi, K+2*hi+1) of row `arow`
    const float* wp = W + (long long)arow * C + 2 * hi;
    // B: lane l16 = column (nb0 + t*16 + l16), rows K+2*hi / K+2*hi+1
    const float* xp = Xb + (long long)(2 * hi) * P + nb0 + l16;
    const long long xinc = 4LL * P;

    v8f acc[4];
    #pragma unroll
    for (int t = 0; t < 4; ++t) acc[t] = (v8f){0.f,0.f,0.f,0.f,0.f,0.f,0.f,0.f};

    #pragma unroll 2
    for (int k = 0; k < C; k += 4) {
        v2f a;
        a.x = wp[0]; a.y = wp[1]; wp += 4;

        v2f b0, b1, b2, b3;
        b0.x = xp[ 0]; b0.y = xp[P +  0];
        b1.x = xp[16]; b1.y = xp[P + 16];
        b2.x = xp[32]; b2.y = xp[P + 32];
        b3.x = xp[48]; b3.y = xp[P + 48];
        xp += xinc;

        acc[0] = __builtin_amdgcn_wmma_f32_16x16x4_f32(false, a, false, b0, (short)0, acc[0], false, false);
        acc[1] = __builtin_amdgcn_wmma_f32_16x16x4_f32(false, a, false, b1, (short)0, acc[1], false, false);
        acc[2] = __builtin_amdgcn_wmma_f32_16x16x4_f32(false, a, false, b2, (short)0, acc[2], false, false);
        acc[3] = __builtin_amdgcn_wmma_f32_16x16x4_f32(false, a, false, b3, (short)0, acc[3], false, false);
    }

    // C/D layout: VGPR v -> row mb+v (lanes 0-15) / mb+v+8 (lanes 16-31)
    #pragma unroll
    for (int v = 0; v < 8; ++v) {
        const int o = mb + v + 8 * hi;
        if (o < O) {
            const float s  = g  ? g[o] * gmul : 1.0f;
            const float bi = bv ? bv[o]       : 0.0f;
            float* yo = Yb + (long long)o * P + nb0 + l16;
            #pragma unroll
            for (int t = 0; t < 4; ++t) {
                float val = acc[t][v] * s + bi;
                if (relu) val = fmaxf(val, 0.0f);
                yo[t * 16] = val;
            }
        }
    }
}

// ---------------------------------------------------------------------------
// Bilinear resize (half-pixel centers, edge clamp) == jax.image.resize upsample
// ---------------------------------------------------------------------------
__global__ void bilinear_kernel(const float* __restrict__ src, float* __restrict__ dst,
                                int C, int iH, int iW, int oH, int oW,
                                long long sBatch, long long dBatch, int nB)
{
    int idx = blockIdx.x * blockDim.x + threadIdx.x;
    int total = nB * C * oH * oW;
    if (idx >= total) return;
    int ow = idx % oW; int t = idx / oW;
    int oh = t % oH;   t /= oH;
    int c  = t % C;    int b = t / C;

    float fy = (oh + 0.5f) * (float)iH / (float)oH - 0.5f;
    float fx = (ow + 0.5f) * (float)iW / (float)oW - 0.5f;
    int y0 = (int)floorf(fy); float wy = fy - (float)y0;
    int x0 = (int)floorf(fx); float wx = fx - (float)x0;
    int y1 = y0 + 1, x1 = x0 + 1;
    y0 = max(0, min(iH - 1, y0)); y1 = max(0, min(iH - 1, y1));
    x0 = max(0, min(iW - 1, x0)); x1 = max(0, min(iW - 1, x1));

    const float* s = src + (long long)b * sBatch + (long long)c * iH * iW;
    float v = (1.f - wy) * ((1.f - wx) * s[y0 * iW + x0] + wx * s[y0 * iW + x1])
            +        wy  * ((1.f - wx) * s[y1 * iW + x0] + wx * s[y1 * iW + x1]);
    dst[(long long)b * dBatch + (long long)c * oH * oW + oh * oW + ow] = v;
}

// ---------------------------------------------------------------------------
// Global average pool over 32x32 per (b,c)
// ---------------------------------------------------------------------------
__global__ void gap_kernel(const float* __restrict__ feat, float* __restrict__ gmean)
{
    int i = blockIdx.x * blockDim.x + threadIdx.x;     // b*512 + c
    if (i >= 2 * 512) return;
    const float* s = feat + (long long)i * 1024;
    float acc = 0.f;
    for (int p = 0; p < 1024; ++p) acc += s[p];
    gmean[i] = acc * (1.0f / 1024.0f);
}

// pool 1x1 conv + BN + relu on the pooled vector (512x512, P=1 per batch)
__global__ void pool_kernel(const float* __restrict__ W, const float* __restrict__ gmean,
                            const float* __restrict__ g, const float* __restrict__ bv,
                            float* __restrict__ poolv)
{
    int i = blockIdx.x * blockDim.x + threadIdx.x;     // b*512 + o
    if (i >= 1024) return;
    int b = i >> 9, o = i & 511;
    const float* x = gmean + b * 512;
    const float* w = W + (long long)o * 512;
    float acc = 0.f;
    for (int c = 0; c < 512; ++c) acc += w[c] * x[c];
    poolv[i] = fmaxf(acc * g[o] * BN_INV + bv[o], 0.0f);
}

// broadcast pooled vector into cat buffer channels [512,1024)
__global__ void bcast_kernel(const float* __restrict__ poolv, float* __restrict__ cat)
{
    int idx = blockIdx.x * blockDim.x + threadIdx.x;
    if (idx >= 2 * 512 * 1024) return;
    int p = idx & 1023; int t = idx >> 10;
    int o = t & 511;    int b = t >> 9;
    cat[((long long)b * 2560 + 512 + o) * 1024 + p] = poolv[b * 512 + o];
}

// ---------------------------------------------------------------------------
// Extract 8x8 query patches: qbuf[p][c][t] from feat (p=b*16+ih*4+iw, t=y*8+x)
// ---------------------------------------------------------------------------
__global__ void qextract_kernel(const float* __restrict__ feat, float* __restrict__ qbuf)
{
    int idx = blockIdx.x * blockDim.x + threadIdx.x;
    if (idx >= 32 * 512 * 64) return;
    int t = idx & 63; int r0 = idx >> 6;
    int c = r0 & 511; int p = r0 >> 9;
    int b = p >> 4; int ih = (p >> 2) & 3; int iw = p & 3;
    int y = t >> 3, x = t & 7;
    qbuf[idx] = feat[((long long)b * 512 + c) * 1024 + (ih * 8 + y) * 32 + iw * 8 + x];
}

// unfold(K=8r, stride 8, pad) + rxr block pool 0.5*(max+mean), zero padding
__global__ void ctxpool_kernel(const float* __restrict__ feat, float* __restrict__ ctx,
                               int r, int pad)
{
    int idx = blockIdx.x * blockDim.x + threadIdx.x;
    if (idx >= 32 * 512 * 64) return;
    int t = idx & 63; int r0 = idx >> 6;
    int c = r0 & 511; int p = r0 >> 9;
    int b = p >> 4; int ih = (p >> 2) & 3; int iw = p & 3;
    int gy0 = ih * 8 - pad + (t >> 3) * r;
    int gx0 = iw * 8 - pad + (t & 7) * r;
    const float* f = feat + ((long long)b * 512 + c) * 1024;
    float mx = -3.4e38f, sum = 0.f;
    for (int dy = 0; dy < r; ++dy)
        for (int dx = 0; dx < r; ++dx) {
            int gy = gy0 + dy, gx = gx0 + dx;
            float v = (gy >= 0 && gy < 32 && gx >= 0 && gx < 32) ? f[gy * 32 + gx] : 0.0f;
            mx = fmaxf(mx, v); sum += v;
        }
    ctx[idx] = 0.5f * (mx + sum / (float)(r * r));
}

// layernorm over channels at each (patch, token): ctx[p][c][t] in-place
__global__ void ln_kernel(float* __restrict__ ctx, const float* __restrict__ g,
                          const float* __restrict__ bv)
{
    int idx = blockIdx.x * blockDim.x + threadIdx.x;
    if (idx >= 32 * 64) return;
    int t = idx & 63; int p = idx >> 6;
    float* base = ctx + (long long)p * 512 * 64 + t;
    float mu = 0.f;
    for (int c = 0; c < 512; ++c) mu += base[c * 64];
    mu *= (1.0f / 512.0f);
    float var = 0.f;
    for (int c = 0; c < 512; ++c) { float d = base[c * 64] - mu; var += d * d; }
    var *= (1.0f / 512.0f);
    float inv = rsqrtf(var + EPS_LN);
    for (int c = 0; c < 512; ++c)
        base[c * 64] = (base[c * 64] - mu) * inv * g[c] + bv[c];
}

// ---------------------------------------------------------------------------
// Per-(patch,head) attention: 64 q tokens x 64 kv tokens, softmax, AV.
// qp[p][512][64], kv[p][1024][64] (k rows [0,512), v rows [512,1024))
// ---------------------------------------------------------------------------
__global__ void attn_kernel(const float* __restrict__ qp, const float* __restrict__ kv,
                            float* __restrict__ out, int hd)
{
    __shared__ float smem[64 * 64];
    const int h = blockIdx.x, p = blockIdx.y;
    const int t = threadIdx.x;                       // 0..63
    const float* q = qp + ((long long)p * 512  + h * hd) * 64;
    const float* k = kv + ((long long)p * 1024 + h * hd) * 64;
    const float* v = kv + ((long long)p * 1024 + 512 + h * hd) * 64;
    const float scale = rsqrtf((float)hd);
    float* row = smem + t * 64;

    float mx = -3.4e38f;
    for (int u = 0; u < 64; ++u) {
        float s = 0.f;
        for (int d = 0; d < hd; ++d) s += q[d * 64 + t] * k[d * 64 + u];
        s *= scale;
        row[u] = s; mx = fmaxf(mx, s);
    }
    float sum = 0.f;
    for (int u = 0; u < 64; ++u) { float e = __expf(row[u] - mx); row[u] = e; sum += e; }
    float inv = 1.0f / sum;

    float* o = out + ((long long)p * 512 + h * hd) * 64;
    for (int d = 0; d < hd; ++d) {
        float acc = 0.f;
        for (int u = 0; u < 64; ++u) acc += row[u] * v[d * 64 + u];
        o[d * 64 + t] = acc * inv;
    }
}

// scatter per-patch proj output back to (B, C, 32, 32) inside cat buffer
__global__ void scatter_kernel(const float* __restrict__ proj, float* __restrict__ cat,
                               int chOff)
{
    int idx = blockIdx.x * blockDim.x + threadIdx.x;
    if (idx >= 32 * 512 * 64) return;
    int t = idx & 63; int r0 = idx >> 6;
    int c = r0 & 511; int p = r0 >> 9;
    int b = p >> 4; int ih = (p >> 2) & 3; int iw = p & 3;
    int y = t >> 3, x = t & 7;
    cat[((long long)b * 2560 + chOff + c) * 1024 + (ih * 8 + y) * 32 + iw * 8 + x] = proj[idx];
}

// ---------------------------------------------------------------------------

static inline void launch_gemm(hipStream_t s, const float* W, const float* X, float* Y,
                               const float* g, const float* bv, float gmul, int relu,
                               int O, int C, int P, long long xs, long long ys, int nb)
{
    dim3 grid((unsigned)(P / 64), (unsigned)((O + 15) / 16), (unsigned)nb);
    gemm_wmma_f32<<<grid, 32, 0, s>>>(W, X, Y, g, bv, gmul, relu, O, C, P, xs, ys);
}

extern "C" void kernel_launch(void* const* d_in, const int* in_sizes, int n_in,
                              void* d_out, int out_size, void* d_ws, size_t ws_size,
                              hipStream_t stream)
{
    (void)in_sizes; (void)n_in; (void)out_size; (void)ws_size;
    const float* f1 = (const float*)d_in[0];   // (2,64,64,64)
    const float* f2 = (const float*)d_in[1];   // (2,128,32,32)
    const float* f3 = (const float*)d_in[2];   // (2,320,16,16)
    const float* f4 = (const float*)d_in[3];   // (2,512,8,8)
    // params in dict insertion order
    enum { C1W=4,C1B,C2W,C2B,C3W,C3B,C4W,C4B, FUW,FUG,FUB, SHW,SHG,SHB,
           POW,POG,POB, CAW,CAG,CAB, LOW,LOG,LOB, PRW,PRB, R0=29 };
    auto PF = [&](int i){ return (const float*)d_in[i]; };

    float* ws = (float*)d_ws;
    long long off = 0;
    auto alloc = [&](long long n){ float* p = ws + off; off += n; return p; };
    float* c4m     = alloc(2LL * 512 * 64);
    float* c3m     = alloc(2LL * 512 * 256);
    float* fusebuf = alloc(2LL * 1536 * 1024);
    float* feat    = alloc(2LL * 512 * 1024);
    float* catbuf  = alloc(2LL * 2560 * 1024);
    float* gmean   = alloc(1024);
    float* poolv   = alloc(1024);
    float* qbuf    = alloc(32LL * 512 * 64);
    float* ctxbuf  = alloc(32LL * 512 * 64);
    float* kvbuf   = alloc(32LL * 1024 * 64);
    float* qproj   = alloc(32LL * 512 * 64);
    float* attnb   = alloc(32LL * 512 * 64);
    float* projb   = alloc(32LL * 512 * 64);
    float* outbuf  = alloc(2LL * 512 * 1024);
    float* lowbuf  = alloc(2LL * 560 * 4096);
    float* fusedb  = alloc(2LL * 512 * 4096);

    const int T = 256;
    auto nblk = [&](long long n){ return (unsigned)((n + T - 1) / T); };

    // --- c2/c3/c4 mlps; concat order in fuse input: [c4, c3, c2] ---
    launch_gemm(stream, PF(C2W), f2, fusebuf + 1024LL * 1024, nullptr, PF(C2B), 1.f, 0,
                512, 128, 1024, 128LL * 1024, 1536LL * 1024, 2);
    launch_gemm(stream, PF(C3W), f3, c3m, nullptr, PF(C3B), 1.f, 0,
                512, 320, 256, 320LL * 256, 512LL * 256, 2);
    launch_gemm(stream, PF(C4W), f4, c4m, nullptr, PF(C4B), 1.f, 0,
                512, 512, 64, 512LL * 64, 512LL * 64, 2);
    bilinear_kernel<<<nblk(2LL*512*1024), T, 0, stream>>>(c3m, fusebuf + 512LL * 1024,
        512, 16, 16, 32, 32, 512LL * 256, 1536LL * 1024, 2);
    bilinear_kernel<<<nblk(2LL*512*1024), T, 0, stream>>>(c4m, fusebuf,
        512, 8, 8, 32, 32, 512LL * 64, 1536LL * 1024, 2);

    // --- fuse conv+bn+relu -> feat ---
    launch_gemm(stream, PF(FUW), fusebuf, feat, PF(FUG), PF(FUB), BN_INV, 1,
                512, 1536, 1024, 1536LL * 1024, 512LL * 1024, 2);

    // --- feat_short -> cat[0:512) ---
    launch_gemm(stream, PF(SHW), feat, catbuf, PF(SHG), PF(SHB), BN_INV, 1,
                512, 512, 1024, 512LL * 1024, 2560LL * 1024, 2);

    // --- pool branch -> cat[512:1024) ---
    gap_kernel<<<nblk(1024), T, 0, stream>>>(feat, gmean);
    pool_kernel<<<nblk(1024), T, 0, stream>>>(PF(POW), gmean, PF(POG), PF(POB), poolv);
    bcast_kernel<<<nblk(2LL*512*1024), T, 0, stream>>>(poolv, catbuf);

    // --- lawin attention branches ---
    qextract_kernel<<<nblk(32LL*512*64), T, 0, stream>>>(feat, qbuf);
    const int ratios[3] = {8, 4, 2};
    const int pads[3]   = {28, 12, 4};
    const int hds[3]    = {8, 32, 128};
    const int headsA[3] = {64, 16, 4};
    for (int i = 0; i < 3; ++i) {
        const int base = R0 + 6 * i;  // q_w, kv_w, p_w, p_b, ds_g, ds_b
        ctxpool_kernel<<<nblk(32LL*512*64), T, 0, stream>>>(feat, ctxbuf, ratios[i], pads[i]);
        ln_kernel<<<nblk(32*64), T, 0, stream>>>(ctxbuf, PF(base + 4), PF(base + 5));
        launch_gemm(stream, PF(base + 0), qbuf, qproj, nullptr, nullptr, 1.f, 0,
                    512, 512, 64, 512LL * 64, 512LL * 64, 32);
        launch_gemm(stream, PF(base + 1), ctxbuf, kvbuf, nullptr, nullptr, 1.f, 0,
                    1024, 512, 64, 512LL * 64, 1024LL * 64, 32);
        attn_kernel<<<dim3(headsA[i], 32), 64, 0, stream>>>(qproj, kvbuf, attnb, hds[i]);
        launch_gemm(stream, PF(base + 2), attnb, projb, nullptr, PF(base + 3), 1.f, 0,
                    512, 512, 64, 512LL * 64, 512LL * 64, 32);
        scatter_kernel<<<nblk(32LL*512*64), T, 0, stream>>>(projb, catbuf, 1024 + 512 * i);
    }

    // --- cat conv+bn+relu -> outbuf ---
    launch_gemm(stream, PF(CAW), catbuf, outbuf, PF(CAG), PF(CAB), BN_INV, 1,
                512, 2560, 1024, 2560LL * 1024, 512LL * 1024, 2);

    // --- low-level fusion: [resize(out,64), c1] ---
    bilinear_kernel<<<nblk(2LL*512*4096), T, 0, stream>>>(outbuf, lowbuf,
        512, 32, 32, 64, 64, 512LL * 1024, 560LL * 4096, 2);
    launch_gemm(stream, PF(C1W), f1, lowbuf + 512LL * 4096, nullptr, PF(C1B), 1.f, 0,
                48, 64, 4096, 64LL * 4096, 560LL * 4096, 2);
    launch_gemm(stream, PF(LOW), lowbuf, fusedb, PF(LOG), PF(LOB), BN_INV, 1,
                512, 560, 4096, 560LL * 4096, 512LL * 4096, 2);

    // --- prediction head ---
    launch_gemm(stream, PF(PRW), fusedb, (float*)d_out, nullptr, PF(PRB), 1.f, 0,
                19, 512, 4096, 512LL * 4096, 19LL * 4096, 2);
}